// NodeRNN_80857054314873
// MI455X (gfx1250) — compile-verified
//
#include <hip/hip_runtime.h>
#include <hip/hip_bf16.h>
#include <math.h>

// ---------------------------------------------------------------------------
// NodeRNN fused pipeline for MI455X (gfx1250, wave32).
//   K1: emb + 3 SE-residual blocks via v_wmma_f32_16x16x4_f32 -> concat (896)
//   K2: gates GEMV  (8192 x 2944, 96.5 MB streamed; HBM-bound ~4.1 us floor)
//   K3: LSTM pointwise -> h_1, c_1
//   K4: decoder (2 x 2048 dot + tanh)
// ---------------------------------------------------------------------------

typedef __attribute__((ext_vector_type(2))) float v2f;
typedef __attribute__((ext_vector_type(8))) float v8f;

#define EPS 1e-5f
#define LDSW 132   // LDS row stride (128 + 4) to avoid 64-bank conflicts

__device__ __forceinline__ float sigmf(float x) { return 1.0f / (1.0f + __expf(-x)); }

// ---------------------------------------------------------------------------
// Kernel A: preamble. One workgroup of 256 threads (8 wave32s).
// Each wave owns a 16-wide output-channel tile; M (batch=6) padded to 16.
// ---------------------------------------------------------------------------
__global__ __launch_bounds__(256) void nodernn_preamble(
    const float* __restrict__ node_pos, const float* __restrict__ edges,
    const float* __restrict__ emb_w,    const float* __restrict__ emb_b,
    const float* __restrict__ conv1_w,
    const float* __restrict__ bn1_g, const float* __restrict__ bn1_b,
    const float* __restrict__ bn1_m, const float* __restrict__ bn1_v,
    const float* __restrict__ conv2_w,
    const float* __restrict__ bn2_g, const float* __restrict__ bn2_b,
    const float* __restrict__ bn2_m, const float* __restrict__ bn2_v,
    const float* __restrict__ se_w1, const float* __restrict__ se_w2,
    float* __restrict__ concat_out)
{
    __shared__ float X [16 * LDSW];   // residual / current activations
    __shared__ float T1[16 * LDSW];   // relu(bn1(conv1(x)))
    __shared__ float T2[16 * LDSW];   // bn2(conv2(t1))
    __shared__ float Hse[6 * 8];      // SE hidden

    const int tid  = threadIdx.x;
    const int lane = tid & 31;
    const int wave = tid >> 5;        // 0..7 -> output-channel tile
    const int g    = lane >> 4;       // half-wave group (K / row-half select)
    const int m    = lane & 15;       // A-row for this lane
    const int n    = wave * 16 + (lane & 15);  // output channel for B/C/D

    // ---- init LDS: rows >= 6 zero, rows < 6 = edges (6 x 128) ----
    for (int i = tid; i < 16 * 128; i += 256) {
        int r = i >> 7, c = i & 127;
        float v = (r < 6) ? edges[r * 128 + c] : 0.0f;
        X [r * LDSW + c] = v;
        T1[r * LDSW + c] = 0.0f;
        T2[r * LDSW + c] = 0.0f;
    }
    __syncthreads();

    for (int blk = 0; blk < 3; ++blk) {
        // per-lane BN constants for this lane's output channel
        const int ci = blk * 128 + n;
        const float s1 = bn1_g[ci] * __frsqrt_rn(bn1_v[ci] + EPS);
        const float m1 = bn1_m[ci], a1 = bn1_b[ci];
        const float s2 = bn2_g[ci] * __frsqrt_rn(bn2_v[ci] + EPS);
        const float m2 = bn2_m[ci], a2 = bn2_b[ci];

        // ---- conv1 (center tap only) : D = X(16x128) * W1^T tile ----
        {
            v8f acc = {};
            const float* wbase = conv1_w + ((size_t)(blk * 128 + n) * 128) * 3 + 1;
            #pragma unroll 4
            for (int k0 = 0; k0 < 128; k0 += 4) {
                const int k = k0 + 2 * g;
                v2f a, b;
                a.x = X[m * LDSW + k];
                a.y = X[m * LDSW + k + 1];
                b.x = wbase[(size_t)k * 3];        // W[n][k][tap=1]
                b.y = wbase[(size_t)(k + 1) * 3];  // W[n][k+1][tap=1]
                acc = __builtin_amdgcn_wmma_f32_16x16x4_f32(
                        false, a, false, b, (short)0, acc, false, false);
            }
            #pragma unroll
            for (int r = 0; r < 8; ++r) {
                int row = r + 8 * g;
                if (row < 6) {
                    float t = (acc[r] - m1) * s1 + a1;
                    T1[row * LDSW + n] = fmaxf(t, 0.0f);
                }
            }
        }
        __syncthreads();

        // ---- conv2 : D = T1(16x128) * W2^T tile, bn2, no relu ----
        {
            v8f acc = {};
            const float* wbase = conv2_w + ((size_t)(blk * 128 + n) * 128) * 3 + 1;
            #pragma unroll 4
            for (int k0 = 0; k0 < 128; k0 += 4) {
                const int k = k0 + 2 * g;
                v2f a, b;
                a.x = T1[m * LDSW + k];
                a.y = T1[m * LDSW + k + 1];
                b.x = wbase[(size_t)k * 3];
                b.y = wbase[(size_t)(k + 1) * 3];
                acc = __builtin_amdgcn_wmma_f32_16x16x4_f32(
                        false, a, false, b, (short)0, acc, false, false);
            }
            #pragma unroll
            for (int r = 0; r < 8; ++r) {
                int row = r + 8 * g;
                if (row < 6)
                    T2[row * LDSW + n] = (acc[r] - m2) * s2 + a2;
            }
        }
        __syncthreads();

        // ---- SE squeeze: h[b][j] = relu(sum_i T2[b][i] * w1[j][i]) ----
        if (tid < 48) {
            int b = tid >> 3, j = tid & 7;
            const float* w1 = se_w1 + (size_t)(blk * 8 + j) * 128;
            float s = 0.0f;
            for (int i = 0; i < 128; ++i) s += T2[b * LDSW + i] * w1[i];
            Hse[b * 8 + j] = fmaxf(s, 0.0f);
        }
        __syncthreads();

        // ---- SE excite + residual + relu, update X in place ----
        for (int t = tid; t < 768; t += 256) {
            int b = t >> 7, c = t & 127;
            const float* w2 = se_w2 + (size_t)(blk * 128 + c) * 8;
            float s = 0.0f;
            #pragma unroll
            for (int j = 0; j < 8; ++j) s += Hse[b * 8 + j] * w2[j];
            float y = sigmf(s);
            X[b * LDSW + c] = fmaxf(T2[b * LDSW + c] * y + X[b * LDSW + c], 0.0f);
        }
        __syncthreads();
    }

    // ---- emit concat = [emb(128), e(6*128)] ----
    if (tid < 128) {
        float e = tanhf(node_pos[0] * emb_w[tid * 2 + 0] +
                        node_pos[1] * emb_w[tid * 2 + 1] + emb_b[tid]);
        concat_out[tid] = e;
    }
    for (int t = tid; t < 768; t += 256)
        concat_out[128 + t] = X[(t >> 7) * LDSW + (t & 127)];
}

// ---------------------------------------------------------------------------
// Kernel B: gates GEMV. One wave per output row; b128 coalesced streaming.
// grid = 1024 blocks x 256 threads (8 rows/block) covers 8192 rows.
// ---------------------------------------------------------------------------
__global__ __launch_bounds__(256) void nodernn_gemv(
    const float* __restrict__ w_ih, const float* __restrict__ w_hh,
    const float* __restrict__ b_ih, const float* __restrict__ b_hh,
    const float* __restrict__ h0,   const float* __restrict__ concat,
    float* __restrict__ gates)
{
    __shared__ __align__(16) float vec[2944];   // [concat(896) | h0(2048)]
    const int tid = threadIdx.x;
    for (int i = tid; i < 896;  i += 256) vec[i]        = concat[i];
    for (int i = tid; i < 2048; i += 256) vec[896 + i]  = h0[i];
    __syncthreads();

    const int lane = tid & 31;
    const int row  = blockIdx.x * 8 + (tid >> 5);

    const float4* a4 = (const float4*)(w_ih + (size_t)row * 896);
    const float4* b4 = (const float4*)(w_hh + (size_t)row * 2048);
    const float4* v4 = (const float4*)vec;            // 224 x float4
    const float4* u4 = (const float4*)(vec + 896);    // 512 x float4

    float s = 0.0f;
    #pragma unroll
    for (int j = lane; j < 224; j += 32) {
        float4 w = a4[j], v = v4[j];
        s += w.x * v.x + w.y * v.y + w.z * v.z + w.w * v.w;
    }
    #pragma unroll 4
    for (int j = lane; j < 512; j += 32) {
        float4 w = b4[j], v = u4[j];
        s += w.x * v.x + w.y * v.y + w.z * v.z + w.w * v.w;
    }
    #pragma unroll
    for (int off = 16; off > 0; off >>= 1) s += __shfl_down(s, off, 32);
    if (lane == 0) gates[row] = s + b_ih[row] + b_hh[row];
}

// ---------------------------------------------------------------------------
// Kernel C: LSTM pointwise. d_out layout: [output(2) | h1(2048) | c1(2048)]
// ---------------------------------------------------------------------------
__global__ __launch_bounds__(256) void nodernn_lstm_cell(
    const float* __restrict__ gates, const float* __restrict__ c0,
    float* __restrict__ d_out)
{
    int i = blockIdx.x * 256 + threadIdx.x;
    if (i < 2048) {
        float ig = gates[i];
        float fg = gates[2048 + i];
        float gg = gates[4096 + i];
        float og = gates[6144 + i];
        float c1 = sigmf(fg) * c0[i] + sigmf(ig) * tanhf(gg);
        float h1 = sigmf(og) * tanhf(c1);
        d_out[2 + i]    = h1;
        d_out[2050 + i] = c1;
    }
}

// ---------------------------------------------------------------------------
// Kernel D: decoder, one block: out[k] = tanh(h1 . dec_w[k] + dec_b[k])
// ---------------------------------------------------------------------------
__global__ __launch_bounds__(256) void nodernn_decoder(
    const float* __restrict__ dec_w, const float* __restrict__ dec_b,
    float* __restrict__ d_out)
{
    __shared__ float r0[256], r1[256];
    const int tid = threadIdx.x;
    float s0 = 0.0f, s1 = 0.0f;
    for (int i = tid; i < 2048; i += 256) {
        float h = d_out[2 + i];
        s0 += h * dec_w[i];
        s1 += h * dec_w[2048 + i];
    }
    r0[tid] = s0; r1[tid] = s1;
    __syncthreads();
    for (int st = 128; st > 0; st >>= 1) {
        if (tid < st) { r0[tid] += r0[tid + st]; r1[tid] += r1[tid + st]; }
        __syncthreads();
    }
    if (tid == 0) {
        d_out[0] = tanhf(r0[0] + dec_b[0]);
        d_out[1] = tanhf(r1[0] + dec_b[1]);
    }
}

// ---------------------------------------------------------------------------
extern "C" void kernel_launch(void* const* d_in, const int* in_sizes, int n_in,
                              void* d_out, int out_size, void* d_ws, size_t ws_size,
                              hipStream_t stream) {
    const float* node_pos = (const float*)d_in[0];
    const float* edges    = (const float*)d_in[1];
    const float* h_0      = (const float*)d_in[2];
    const float* c_0      = (const float*)d_in[3];
    const float* emb_w    = (const float*)d_in[4];
    const float* emb_b    = (const float*)d_in[5];
    const float* conv1_w  = (const float*)d_in[6];
    const float* bn1_g    = (const float*)d_in[7];
    const float* bn1_b    = (const float*)d_in[8];
    const float* bn1_m    = (const float*)d_in[9];
    const float* bn1_v    = (const float*)d_in[10];
    const float* conv2_w  = (const float*)d_in[11];
    const float* bn2_g    = (const float*)d_in[12];
    const float* bn2_b    = (const float*)d_in[13];
    const float* bn2_m    = (const float*)d_in[14];
    const float* bn2_v    = (const float*)d_in[15];
    const float* se_w1    = (const float*)d_in[16];
    const float* se_w2    = (const float*)d_in[17];
    const float* w_ih     = (const float*)d_in[18];
    const float* w_hh     = (const float*)d_in[19];
    const float* b_ih     = (const float*)d_in[20];
    const float* b_hh     = (const float*)d_in[21];
    const float* dec_w    = (const float*)d_in[22];
    const float* dec_b    = (const float*)d_in[23];

    float* out    = (float*)d_out;
    float* ws     = (float*)d_ws;
    float* concat = ws;           // 896 floats
    float* gates  = ws + 1024;    // 8192 floats

    nodernn_preamble<<<1, 256, 0, stream>>>(
        node_pos, edges, emb_w, emb_b,
        conv1_w, bn1_g, bn1_b, bn1_m, bn1_v,
        conv2_w, bn2_g, bn2_b, bn2_m, bn2_v,
        se_w1, se_w2, concat);

    nodernn_gemv<<<1024, 256, 0, stream>>>(
        w_ih, w_hh, b_ih, b_hh, h_0, concat, gates);

    nodernn_lstm_cell<<<8, 256, 0, stream>>>(gates, c_0, out);

    nodernn_decoder<<<1, 256, 0, stream>>>(dec_w, dec_b, out);
}